// Strict2_5DLoss_22385369547317
// MI455X (gfx1250) — compile-verified
//
#include <hip/hip_runtime.h>
#include <math.h>

typedef __attribute__((ext_vector_type(2))) float v2f;
typedef __attribute__((ext_vector_type(8))) float v8f;

#define ETA_C   3.0f
#define POS_WC  1.2f
#define KSEL    64
#define BATCH   8
#define NGT     32
#define CAP     2048

// ---------------------------------------------------------------- helpers

__device__ __forceinline__ float softplusf(float x) {
  // stable: max(x,0) + log1p(exp(-|x|))  (matches jax.nn.softplus)
  return fmaxf(x, 0.0f) + log1pf(expf(-fabsf(x)));
}

// Wave32 sum via the matrix pipe: D = A(16x4) * ones(4x16) + 0.
// Lane L<16 holds (M=L,K=0..1); lane L>=16 holds (M=L-16,K=2..3).
// v in a.x covers every lane exactly once; column 0 of D lives in
// lane 0 (rows 0..7) and lane 16 (rows 8..15).
__device__ __forceinline__ float wave_sum_wmma(float v) {
  v2f a; a.x = v;    a.y = 0.0f;
  v2f b; b.x = 1.0f; b.y = 1.0f;
  v8f c = {};
  c = __builtin_amdgcn_wmma_f32_16x16x4_f32(false, a, false, b,
                                            (short)0, c, false, false);
  float s = c[0] + c[1] + c[2] + c[3] + c[4] + c[5] + c[6] + c[7];
  return __shfl(s, 0, 32) + __shfl(s, 16, 32);
}

__device__ __forceinline__ float block_sum(float v, float* smem) {
  float w = wave_sum_wmma(v);
  int lane = threadIdx.x & 31;
  int wid  = threadIdx.x >> 5;
  if (lane == 0) smem[wid] = w;
  __syncthreads();
  float r = 0.0f;
  if (threadIdx.x == 0) {
    int nw = blockDim.x >> 5;
    for (int i = 0; i < nw; ++i) r += smem[i];
  }
  __syncthreads();
  return r; // valid on thread 0
}

// point-to-segment distance with precomputed edge constants;
// w = p - segment_start, v = segment vector, inv = 1/(|v|^2 + 1e-9)
__device__ __forceinline__ float seg_dist_pre(float wx, float wy,
                                              float vx, float vy, float inv) {
  float t = fminf(fmaxf((wx * vx + wy * vy) * inv, 0.0f), 1.0f);
  float dx = wx - t * vx;
  float dy = wy - t * vy;
  return sqrtf(dx * dx + dy * dy + 1e-12f);
}

// ---------------------------------------------------------------- kernels

__global__ void zero_acc_kernel(float* acc) {
  if (threadIdx.x < 8) acc[threadIdx.x] = 0.0f;
}

__global__ void zero_words_kernel(unsigned int* p, int n) {
  int s = gridDim.x * blockDim.x;
  for (int i = blockIdx.x * blockDim.x + threadIdx.x; i < n; i += s) p[i] = 0u;
}

// One workgroup per (b,g): collect positive anchors; select 64 smallest.
// All downstream loss terms are order-invariant sums over the selected set,
// so the bitonic sort is only needed when count > KSEL, and then only over
// the next power-of-two >= count.
__global__ void topk_select_kernel(const float* __restrict__ gt,
                                   int* __restrict__ selIdx,
                                   unsigned char* __restrict__ mask,
                                   int HW, int wShift, float stride) {
  __shared__ unsigned long long keys[CAP];
  __shared__ int cnt;
  const int tid   = threadIdx.x;
  const int bg    = blockIdx.x;          // b*NGT + g
  const int b     = bg / NGT;
  const int wMask = (1 << wShift) - 1;

  for (int i = tid; i < CAP; i += blockDim.x) keys[i] = ~0ULL;
  if (tid == 0) cnt = 0;
  __syncthreads();

  const float* tp = gt + (size_t)bg * 6;
  const float ax = tp[0], ay = tp[1];
  const float bx = tp[2], by = tp[3];
  const float cx = tp[4], cy = tp[5];
  // edge vectors + inverse squared lengths (SGPR-resident constants)
  const float e1x = bx - ax, e1y = by - ay;
  const float e2x = cx - bx, e2y = cy - by;
  const float e3x = ax - cx, e3y = ay - cy;
  const float i1 = 1.0f / (e1x * e1x + e1y * e1y + 1e-9f);
  const float i2 = 1.0f / (e2x * e2x + e2y * e2y + 1e-9f);
  const float i3 = 1.0f / (e3x * e3x + e3y * e3y + 1e-9f);

  for (int hw = tid; hw < HW; hw += blockDim.x) {
    int iy = hw >> wShift, ix = hw & wMask;
    float px = ((float)ix + 0.5f) * stride;
    float py = ((float)iy + 0.5f) * stride;
    float d1 = (px - bx) * (ay - by) - (ax - bx) * (py - by);
    float d2 = (px - cx) * (by - cy) - (bx - cx) * (py - cy);
    float d3 = (px - ax) * (cy - ay) - (cx - ax) * (py - ay);
    bool hneg = (d1 < 0.0f) | (d2 < 0.0f) | (d3 < 0.0f);
    bool hpos = (d1 > 0.0f) | (d2 > 0.0f) | (d3 > 0.0f);
    bool inside = !(hneg && hpos);
    float dist = fminf(seg_dist_pre(px - ax, py - ay, e1x, e1y, i1),
                 fminf(seg_dist_pre(px - bx, py - by, e2x, e2y, i2),
                       seg_dist_pre(px - cx, py - cy, e3x, e3y, i3)));
    if (inside || dist <= ETA_C) {
      int slot = atomicAdd(&cnt, 1);                 // ds_add_rtn_u32
      if (slot < CAP)
        keys[slot] = ((unsigned long long)__float_as_uint(dist) << 32)
                   | (unsigned long long)(unsigned int)hw;
    }
  }
  __syncthreads();

  const int total = cnt < CAP ? cnt : CAP;          // uniform across block
  if (total > KSEL) {
    int P = 2 * KSEL;                                // sort span: pow2 >= total
    while (P < total) P <<= 1;
    for (int k = 2; k <= P; k <<= 1) {
      for (int j = k >> 1; j > 0; j >>= 1) {
        for (int i = tid; i < P; i += blockDim.x) {
          int ixj = i ^ j;
          if (ixj > i) {
            bool up = ((i & k) == 0);
            unsigned long long a0 = keys[i], b0 = keys[ixj];
            if (up ? (a0 > b0) : (a0 < b0)) { keys[i] = b0; keys[ixj] = a0; }
          }
        }
        __syncthreads();
      }
    }
  }

  const int vcount = total < KSEL ? total : KSEL;
  if (tid < KSEL) {
    int out = -1;
    if (tid < vcount) {
      unsigned int hw = (unsigned int)(keys[tid] & 0xFFFFFFFFULL);
      out = (int)hw;
      mask[(size_t)b * HW + hw] = 1;                 // scatter-max of 1.0
    }
    selIdx[bg * KSEL + tid] = out;
  }
}

// BCE over all anchors of a level + count of positive anchors (pos_now).
// float4/uchar4 streams; N/4 is a multiple of blockDim so EXEC stays full.
__global__ void obj_loss_kernel(const float* __restrict__ obj,
                                const unsigned char* __restrict__ mask,
                                int N, float* __restrict__ acc) {
  __shared__ float smem[8];
  const float4* o4 = (const float4*)obj;
  const uchar4* m4 = (const uchar4*)mask;
  const int n4 = N >> 2;
  float objsum = 0.0f, posnow = 0.0f;
  int stride = gridDim.x * blockDim.x;
  for (int i = blockIdx.x * blockDim.x + threadIdx.x; i < n4; i += stride) {
    __builtin_prefetch(&o4[i + stride]);             // global_prefetch_b8
    float4 x = o4[i];
    uchar4 m = m4[i];
    float t0 = (float)m.x, t1 = (float)m.y, t2 = (float)m.z, t3 = (float)m.w;
    objsum += POS_WC * t0 * softplusf(-x.x) + (1.0f - t0) * softplusf(x.x);
    objsum += POS_WC * t1 * softplusf(-x.y) + (1.0f - t1) * softplusf(x.y);
    objsum += POS_WC * t2 * softplusf(-x.z) + (1.0f - t2) * softplusf(x.z);
    objsum += POS_WC * t3 * softplusf(-x.w) + (1.0f - t3) * softplusf(x.w);
    posnow += t0 + t1 + t2 + t3;
  }
  float bo = block_sum(objsum, smem);
  float bp = block_sum(posnow, smem);
  if (threadIdx.x == 0) {
    atomicAdd(acc + 1, bo);   // obj_s
    atomicAdd(acc + 4, bp);   // pos_now (for neg)
  }
}

// One thread per selected (b,g,k) entry: cls BCE + p0 + 2x2 chamfer.
__global__ void cls_reg_kernel(const float* __restrict__ reg,
                               const float* __restrict__ cls,
                               const float* __restrict__ gt,
                               const int* __restrict__ selIdx,
                               int HW, int wShift, float stride,
                               float* __restrict__ acc) {
  __shared__ float smem[8];
  const int gid = blockIdx.x * blockDim.x + threadIdx.x;  // < BATCH*NGT*KSEL
  const int b   = gid / (NGT * KSEL);
  const int bg  = gid / KSEL;
  const int wMask = (1 << wShift) - 1;
  int idx = selIdx[gid];
  float regc = 0.0f, clsc = 0.0f, posc = 0.0f;
  if (idx >= 0) {
    posc = 1.0f;
    clsc = softplusf(-cls[(size_t)b * HW + idx]);
    int iy = idx >> wShift, ix = idx & wMask;
    float ancx = ((float)ix + 0.5f) * stride;
    float ancy = ((float)iy + 0.5f) * stride;
    float inv  = 1.0f / stride;
    const float* tp = gt + (size_t)bg * 6;
    float g0x = (tp[0] - ancx) * inv, g0y = (tp[1] - ancy) * inv;
    float g1x = (tp[2] - ancx) * inv, g1y = (tp[3] - ancy) * inv;
    float g2x = (tp[4] - ancx) * inv, g2y = (tp[5] - ancy) * inv;
    const float* rb = reg + (size_t)b * 6 * HW + idx;
    float p[6];
#pragma unroll
    for (int c = 0; c < 6; ++c) {
      float v = rb[(size_t)c * HW];
      p[c] = fminf(fmaxf(v, -64.0f), 64.0f);
    }
    float dx0 = p[0] - g0x, dy0 = p[1] - g0y;
    float p0 = dx0 * dx0 + dy0 * dy0;
    float dm00 = sqrtf((p[2]-g1x)*(p[2]-g1x) + (p[3]-g1y)*(p[3]-g1y));
    float dm01 = sqrtf((p[2]-g2x)*(p[2]-g2x) + (p[3]-g2y)*(p[3]-g2y));
    float dm10 = sqrtf((p[4]-g1x)*(p[4]-g1x) + (p[5]-g1y)*(p[5]-g1y));
    float dm11 = sqrtf((p[4]-g2x)*(p[4]-g2x) + (p[5]-g2y)*(p[5]-g2y));
    float cd = fminf(dm00, dm01) + fminf(dm10, dm11)
             + fminf(dm00, dm10) + fminf(dm01, dm11);
    regc = p0 + cd;   // L_P0 = L_CD = 1
  }
  float br = block_sum(regc, smem);
  float bc = block_sum(clsc, smem);
  float bp = block_sum(posc, smem);
  if (threadIdx.x == 0) {
    atomicAdd(acc + 0, br);   // reg_s
    atomicAdd(acc + 2, bc);   // cls_s
    atomicAdd(acc + 3, bp);   // pos
  }
}

__global__ void finalize_kernel(const float* __restrict__ acc,
                                float* __restrict__ out) {
  if (threadIdx.x == 0 && blockIdx.x == 0) {
    float pos = acc[3];
    float neg = 688128.0f - acc[4];   // sum_l B*HW_l = 8*(65536+16384+4096)
    float pos_eps = fmaxf(pos, 1.0f);
    float neg_c   = fmaxf(neg, 1.0f);
    float total = acc[0] / pos_eps + acc[1] / (pos_eps + neg_c)
                + acc[2] / pos_eps;
    out[0] = isfinite(total) ? total : 0.0f;
  }
}

// ---------------------------------------------------------------- launcher

extern "C" void kernel_launch(void* const* d_in, const int* in_sizes, int n_in,
                              void* d_out, int out_size, void* d_ws, size_t ws_size,
                              hipStream_t stream) {
  (void)in_sizes; (void)n_in; (void)out_size; (void)ws_size;
  const float* regs[3] = {(const float*)d_in[0], (const float*)d_in[3], (const float*)d_in[6]};
  const float* objs[3] = {(const float*)d_in[1], (const float*)d_in[4], (const float*)d_in[7]};
  const float* clss[3] = {(const float*)d_in[2], (const float*)d_in[5], (const float*)d_in[8]};
  const float* gt      =  (const float*)d_in[9];

  float*         acc    = (float*)d_ws;
  int*           selIdx = (int*)((char*)d_ws + 256);
  unsigned char* mask   = (unsigned char*)((char*)d_ws + 256
                          + (size_t)(BATCH * NGT * KSEL) * sizeof(int));

  const int   SH[3] = {8, 7, 6};          // log2(Ws) per level
  const float S[3]  = {8.0f, 16.0f, 32.0f};

  zero_acc_kernel<<<1, 32, 0, stream>>>(acc);
  for (int l = 0; l < 3; ++l) {
    int Ws = 1 << SH[l];
    int HW = Ws * Ws;
    int N  = BATCH * HW;
    zero_words_kernel<<<64, 256, 0, stream>>>((unsigned int*)mask, N / 4);
    topk_select_kernel<<<BATCH * NGT, 256, 0, stream>>>(gt, selIdx, mask,
                                                        HW, SH[l], S[l]);
    obj_loss_kernel<<<256, 256, 0, stream>>>(objs[l], mask, N, acc);
    cls_reg_kernel<<<(BATCH * NGT * KSEL) / 256, 256, 0, stream>>>(
        regs[l], clss[l], gt, selIdx, HW, SH[l], S[l], acc);
  }
  finalize_kernel<<<1, 1, 0, stream>>>(acc, (float*)d_out);
}